// differN_new_conv_21285857919124
// MI455X (gfx1250) — compile-verified
//
#include <hip/hip_runtime.h>
#include <math.h>

typedef __attribute__((ext_vector_type(2))) float v2f;
typedef __attribute__((ext_vector_type(8))) float v8f;

#define WIRES 10
#define QDIM 1024
#define NQW 720   // qweights floats: 3*4*2*10*3

// Rot(phi,theta,omega) = RZ(omega) RY(theta) RZ(phi), as 4 complex entries.
__device__ __forceinline__ void rot_mat(float phi, float th, float om,
                                        float* m) {
  const float hpo = 0.5f * (phi + om), hmo = 0.5f * (phi - om);
  const float ct = __cosf(0.5f * th), st = __sinf(0.5f * th);
  const float cpo = __cosf(hpo), spo = __sinf(hpo);
  const float cmo = __cosf(hmo), smo = __sinf(hmo);
  m[0] =  cpo * ct; m[1] = -spo * ct;   // m00
  m[2] = -cmo * st; m[3] = -smo * st;   // m01
  m[4] =  cmo * st; m[5] = -smo * st;   // m10
  m[6] =  cpo * ct; m[7] =  spo * ct;   // m11
}

// ---------------------------------------------------------------------------
// Single fused kernel: one block (8 wave32) per batch element.
//  Prologue: 3x3 stride-2 conv + bias + spatial mean -> 10 angles (in LDS).
//  Body:     3-rep quantum circuit with ping-pong 1024-amplitude complex
//            state in LDS. Adjacent-wire gate pairs are kron'd into an 8x8
//            real matrix and executed as OUT(quads x 8) = V(quads x 8) @ G2^T
//            via two K-chained V_WMMA_F32_16X16X4_F32 per 16-quad tile.
//            Diagonal RZ-phase / CZ-sign factors are folded into the
//            A-fragment loads; each rep's trailing diagonal is dropped
//            (|psi|^2-invariant). One barrier per gate-pair step.
// ---------------------------------------------------------------------------
__global__ __launch_bounds__(256) void qcircuit_fused_kernel(
    const float* __restrict__ x, const float* __restrict__ cw,
    const float* __restrict__ cb, const float* __restrict__ qw,
    float* __restrict__ out) {
  const int b = blockIdx.x;
  const int tid = threadIdx.x;
  const int lane = tid & 31;
  const int wave = tid >> 5;
  const int hi = lane >> 4;    // lane half (selects K slot pair in fragments)
  const int mrow = lane & 15;  // A row within tile / N column of D

  __shared__ float sreb[2][QDIM];
  __shared__ float simb[2][QDIM];
  __shared__ float sqw[NQW];
  __shared__ float sang[WIRES];

  // ===================== conv + mean prologue =====================
  // Stage input tile (4KB) in sreb[0], conv weights in simb[0][0..89].
#pragma unroll
  for (int k = 0; k < 4; ++k)
    sreb[0][tid + 256 * k] = x[b * 1024 + tid + 256 * k];
  if (tid < 90) simb[0][tid] = cw[tid];
  if (tid < WIRES) sang[tid] = 0.f;
  for (int idx = tid; idx < NQW; idx += 256) sqw[idx] = qw[idx];
  __syncthreads();

  {
    const int oy = tid >> 4;
    const int ox = tid & 15;
    float acc[WIRES];
#pragma unroll
    for (int c = 0; c < WIRES; ++c) acc[c] = 0.f;
#pragma unroll
    for (int ky = 0; ky < 3; ++ky) {
      const int iy = oy * 2 - 1 + ky;
      if (iy < 0 || iy >= 32) continue;
#pragma unroll
      for (int kx = 0; kx < 3; ++kx) {
        const int ix = ox * 2 - 1 + kx;
        if (ix < 0 || ix >= 32) continue;
        const float xin = sreb[0][iy * 32 + ix];
#pragma unroll
        for (int c = 0; c < WIRES; ++c)
          acc[c] = fmaf(simb[0][c * 9 + ky * 3 + kx], xin, acc[c]);
      }
    }
    // wave32 shuffle reduce, one LDS float-atomic per wave per channel
#pragma unroll
    for (int c = 0; c < WIRES; ++c) {
      float v = acc[c];
#pragma unroll
      for (int off = 16; off > 0; off >>= 1) v += __shfl_down(v, off, 32);
      if (lane == 0) atomicAdd(&sang[c], v);
    }
    __syncthreads();
    if (tid < WIRES) sang[tid] = sang[tid] * (1.0f / 256.0f) + cb[tid];
    __syncthreads();
  }

  // ===================== quantum circuit =====================
  int cur = 0;
  for (int rep = 0; rep < 3; ++rep) {
    float ang[WIRES];
#pragma unroll
    for (int w = 0; w < WIRES; ++w) ang[w] = sang[w];

    // |0...0> into current buffer (each thread owns its own d's)
#pragma unroll
    for (int k = 0; k < 4; ++k) {
      const int d = tid + 256 * k;
      sreb[cur][d] = (d == 0) ? 1.f : 0.f;
      simb[cur][d] = 0.f;
    }
    __syncthreads();

    for (int i = 0; i < 4; ++i) {
      for (int l = 0; l < 2; ++l) {
        for (int jp = 0; jp < 5; ++jp) {     // wire pairs (0,1)(2,3)...(8,9)
          const int q = 2 * jp;
          // pending diagonals applied during this step's A loads
          const bool doPhase = (jp == 0 && l == 0);
          const int czr = (jp != 0) ? 0 : (l == 1 ? 1 : (i > 0 ? 2 : 0));

          const float* __restrict__ sr = sreb[cur];
          const float* __restrict__ si = simb[cur];
          float* __restrict__ dr = sreb[cur ^ 1];
          float* __restrict__ di = simb[cur ^ 1];

          // ---- wave-uniform kron(Rot_q, Rot_{q+1}) ----
          const int wbase = (((rep * 4 + i) * 2 + l) * WIRES + q) * 3;
          float mq[8], mp[8];
          rot_mat(sqw[wbase + 0], sqw[wbase + 1], sqw[wbase + 2], mq);
          rot_mat(sqw[wbase + 3], sqw[wbase + 4], sqw[wbase + 5], mp);

          // ---- B fragments (8x16 real G2^T split into two K=4 slabs) ----
          const int n = mrow;
          const bool nv = (n < 8);
          const int a = n >> 1;              // complex output index 0..3
          const int arow = (a >> 1) & 1;     // masked: no OOB for n>=8 lanes
          const int acol = a & 1;
          const bool nodd = (n & 1) != 0;
          v2f bf[2];
#pragma unroll
          for (int w = 0; w < 2; ++w) {
#pragma unroll
            for (int j = 0; j < 2; ++j) {
              const int kk = 4 * w + 2 * hi + j;  // global K index 0..7
              const int bcol = kk >> 1;           // complex input index 0..3
              const int klow = kk & 1;
              // G2[a][bcol] = mq[arow][bcol>>1] * mp[acol][bcol&1] (complex)
              const int qi_ = (arow * 2 + (bcol >> 1)) * 2;
              const int pi_ = (acol * 2 + (bcol & 1)) * 2;
              const float qr = mq[qi_], qim = mq[qi_ + 1];
              const float pr = mp[pi_], pim = mp[pi_ + 1];
              const float cr = qr * pr - qim * pim;
              const float ci = qr * pim + qim * pr;
              float val;
              if (!nodd) val = klow ? -ci : cr;   // real-output rows
              else       val = klow ?  cr : ci;   // imag-output rows
              bf[w][j] = nv ? val : 0.f;
            }
          }

          // ---- A fragments with fused diagonals ----
          const int pos = 9 - q;         // bit position of wire q
          const int S0 = 1 << pos;       // stride of wire q
          const int S1 = S0 >> 1;        // stride of wire q+1
          const int tqbase = wave * 32;  // 32 quads per wave, 2 tiles of 16
          v2f af[2][2];
#pragma unroll
          for (int t = 0; t < 2; ++t) {
#pragma unroll
            for (int w = 0; w < 2; ++w) {
              const int ai = 2 * w + hi;                 // quad component 0..3
              const int p = tqbase + t * 16 + mrow;
              const int basei = ((p >> (pos - 1)) << (pos + 1)) | (p & (S1 - 1));
              const int dd = basei + (ai >> 1) * S0 + (ai & 1) * S1;
              float re = sr[dd], im = si[dd];
              if (czr) {  // fused CZ-ring sign: (-1)^popc(d & rotl10(d,r))
                const int f = ((dd << czr) | (dd >> (10 - czr))) & 1023;
                if (__popc(dd & f) & 1) { re = -re; im = -im; }
              }
              if (doPhase) {  // fused data re-upload RZ diagonal
                float ph = 0.f;
#pragma unroll
                for (int wi = 0; wi < WIRES; ++wi)
                  ph = fmaf(ang[wi], (float)((dd >> (9 - wi)) & 1) - 0.5f, ph);
                const float c = __cosf(ph), s = __sinf(ph);
                const float r2 = re * c - im * s;
                im = re * s + im * c;
                re = r2;
              }
              af[t][w][0] = re;
              af[t][w][1] = im;
            }
          }

          // ---- two K-chained WMMAs per tile + scatter to dst buffer ----
#pragma unroll
          for (int t = 0; t < 2; ++t) {
            v8f acc = {};
            acc = __builtin_amdgcn_wmma_f32_16x16x4_f32(
                false, af[t][0], false, bf[0], (short)0, acc, false, false);
            acc = __builtin_amdgcn_wmma_f32_16x16x4_f32(
                false, af[t][1], false, bf[1], (short)0, acc, false, false);
            if (nv) {
#pragma unroll
              for (int v = 0; v < 8; ++v) {
                const int p = tqbase + t * 16 + v + 8 * hi;  // D row layout
                const int basei = ((p >> (pos - 1)) << (pos + 1)) | (p & (S1 - 1));
                const int addr = basei + arow * S0 + acol * S1;
                if (nodd) di[addr] = acc[v];
                else      dr[addr] = acc[v];
              }
            }
          }
          cur ^= 1;
          __syncthreads();  // single barrier per step (ping-pong buffers)
        }  // jp
      }  // l
    }  // i (spectrum)

    // measurement: |amp|^2 is invariant under the dropped trailing diagonal
    if (rep < 2) {
#pragma unroll
      for (int k = 0; k < 4; ++k) {
        const int d = tid + 256 * k;
        if (d < WIRES) {
          const float re = sreb[cur][d], im = simb[cur][d];
          const float p = (re * re + im * im) * 1024.f;
          sang[d] = fminf(fmaxf(p, 0.f), 1.f);
        }
      }
      __syncthreads();
    } else {
#pragma unroll
      for (int k = 0; k < 4; ++k) {
        const int d = tid + 256 * k;
        const float re = sreb[cur][d], im = simb[cur][d];
        const float p = (re * re + im * im) * 1024.f;
        out[b * 1024 + d] = fminf(fmaxf(p, 0.f), 1.f);
      }
    }
  }  // rep
}

// ---------------------------------------------------------------------------
extern "C" void kernel_launch(void* const* d_in, const int* in_sizes, int n_in,
                              void* d_out, int out_size, void* d_ws, size_t ws_size,
                              hipStream_t stream) {
  const float* x      = (const float*)d_in[0];  // (512,1,32,32)
  const float* conv_w = (const float*)d_in[1];  // (10,1,3,3)
  const float* conv_b = (const float*)d_in[2];  // (10,)
  const float* qwts   = (const float*)d_in[3];  // (3,4,2,10,3)
  float* out = (float*)d_out;                   // (512,1,32,32)

  qcircuit_fused_kernel<<<512, 256, 0, stream>>>(x, conv_w, conv_b, qwts, out);
  (void)in_sizes; (void)n_in; (void)out_size; (void)d_ws; (void)ws_size;
}